// ReservoirRNNCell_65377992180032
// MI455X (gfx1250) — compile-verified
//
#include <hip/hip_runtime.h>
#include <hip/hip_bf16.h>
#include <math.h>

// Problem constants from the reference
#define BATCH 16
#define IDIM 1024
#define HDIM 2048

typedef float v2f __attribute__((ext_vector_type(2)));
typedef float v8f __attribute__((ext_vector_type(8)));

// ---------------------------------------------------------------------------
// Kernel 1: input_contrib = x_t @ W_ih_w^T + b using V_WMMA_F32_16X16X4_F32.
// D = A x B: A = x tile (16x4, M x K), B[k,n] = W_ih_w[n,k] tile (4x16).
// One wave per 16x16 output tile; 128 tiles (N=2048), K loop of 256 steps.
// Fragment layout per ISA 7.12.2:
//   A (16x4 f32): lanes 0-15 -> M=lane, v0=K0 v1=K1; lanes 16-31 -> v0=K2 v1=K3
//   B (4x16 f32): lanes index N, v0=K0 row (half0)/K2 row (half1), v1=K1/K3
//   C/D (16x16 f32): VGPR r, lanes 0-15 -> M=r, lanes 16-31 -> M=r+8; N=lane&15
// ---------------------------------------------------------------------------
__global__ __launch_bounds__(256) void ic_gemm_wmma(
    const float* __restrict__ x,     // [B, I]
    const float* __restrict__ Wih,   // [H, I]
    const float* __restrict__ bias,  // [H]
    float* __restrict__ ic)          // [B, H]
{
  const int wave = threadIdx.x >> 5;
  const int lane = threadIdx.x & 31;
  const int tile = blockIdx.x * 8 + wave;   // 0..127
  const int n0   = tile * 16;

  const int half = lane >> 4;               // 0 or 1 (K-half)
  const int idx  = lane & 15;               // M for A, N for B
  const int kk   = half * 2;                // K offset within step of 4

  const float* arow = x   + (size_t)idx * IDIM;          // A row (M = idx)
  const float* brow = Wih + (size_t)(n0 + idx) * IDIM;   // B col n -> W row n

  v8f c = {0.f, 0.f, 0.f, 0.f, 0.f, 0.f, 0.f, 0.f};

  for (int k0 = 0; k0 < IDIM; k0 += 4) {
    const float2 av = *(const float2*)(arow + k0 + kk);  // 8B aligned (kk even)
    const float2 bv = *(const float2*)(brow + k0 + kk);
    v2f a, b;
    a.x = av.x; a.y = av.y;
    b.x = bv.x; b.y = bv.y;
    // 8 args: (neg_a, A, neg_b, B, c_mod, C, reuse_a, reuse_b)
    c = __builtin_amdgcn_wmma_f32_16x16x4_f32(
        false, a, false, b, (short)0, c, false, false);
  }

  const float bv = bias[n0 + idx];
#pragma unroll
  for (int r = 0; r < 8; ++r) {
    const int m = r + half * 8;
    ic[(size_t)m * HDIM + n0 + idx] = c[r] + bv;
  }
}

// ---------------------------------------------------------------------------
// Kernel 2: per-(b,o) Gumbel-softmax over h, fused weighted sum + tanh.
//   z[h]   = W_hh[o,h]*mask[o,h]/tau + ( -ln(ln2) - ln2*log2(-log2 u[b,o,h]) )
//   out    = tanh(ic[b,o] + sum(e^z * h_prev[b,h]) / sum(e^z))
// Constant stability offset (cancels exactly in the ratio) avoids online-max
// rescale exps: 2x v_log_f32 + 1x v_exp_f32 per element — trans-minimal.
// 8 waves/block, one wave per o-row; h_prev[b] staged in LDS (8 KB).
// ---------------------------------------------------------------------------
__global__ __launch_bounds__(256) void gumbel_contrib(
    const float* __restrict__ Whh,    // [H, H]
    const float* __restrict__ mask,   // [H, H]
    const float* __restrict__ u,      // [B, H, H]
    const float* __restrict__ h_prev, // [B, H]
    const float* __restrict__ ic,     // [B, H]
    const float* __restrict__ temp,   // [1]
    float* __restrict__ out)          // [B, H]
{
  __shared__ __align__(16) float hp_lds[HDIM];

  const int b      = blockIdx.x >> 8;          // 256 blocks per batch element
  const int o_base = (blockIdx.x & 255) * 8;

  // Stage h_prev[b] into LDS (coalesced, 8 floats/thread).
  for (int i = threadIdx.x; i < HDIM; i += 256)
    hp_lds[i] = h_prev[(size_t)b * HDIM + i];
  __syncthreads();

  const int wave = threadIdx.x >> 5;
  const int lane = threadIdx.x & 31;
  const int o    = o_base + wave;

  const float inv_tau = 1.0f / fmaxf(temp[0], 1e-3f);

  const size_t urow = ((size_t)b * HDIM + o) * HDIM;
  const float* wr = Whh  + (size_t)o * HDIM;
  const float* mr = mask + (size_t)o * HDIM;

  const float NLN2 = -0.69314718055994531f;  // -ln 2
  const float GC   =  0.36651292058166435f;  // -ln(ln 2)
  const float L2E  =  1.44269504088896340f;  // log2 e
  const float OFF  =  20.0f;                 // constant stability offset

  float s = 0.0f;   // sum exp(z - OFF)
  float t = 0.0f;   // sum exp(z - OFF) * h_prev

  for (int j = lane * 4; j < HDIM; j += 128) {
    const float4 uv = *(const float4*)(u  + urow + j);
    const float4 wv = *(const float4*)(wr + j);
    const float4 mv = *(const float4*)(mr + j);
    const float4 hv = *(const float4*)(&hp_lds[j]);

    const float us[4] = {uv.x, uv.y, uv.z, uv.w};
    const float ws[4] = {wv.x, wv.y, wv.z, wv.w};
    const float ms[4] = {mv.x, mv.y, mv.z, mv.w};
    const float hs[4] = {hv.x, hv.y, hv.z, hv.w};
#pragma unroll
    for (int c = 0; c < 4; ++c) {
      // gumbel = -ln(-ln u) built from two native log2 ops
      const float g = fmaf(NLN2, __log2f(-__log2f(us[c])), GC);
      const float z = fmaf(ws[c] * ms[c], inv_tau, g);
      const float e = __builtin_amdgcn_exp2f((z - OFF) * L2E);
      s += e;
      t  = fmaf(e, hs[c], t);
    }
  }

  // wave32 butterfly reduction
#pragma unroll
  for (int m = 16; m > 0; m >>= 1) {
    s += __shfl_xor(s, m, 32);
    t += __shfl_xor(t, m, 32);
  }

  if (lane == 0) {
    const size_t bo = (size_t)b * HDIM + o;
    out[bo] = tanhf(ic[bo] + t / s);
  }
}

extern "C" void kernel_launch(void* const* d_in, const int* in_sizes, int n_in,
                              void* d_out, int out_size, void* d_ws, size_t ws_size,
                              hipStream_t stream) {
  const float* x_t     = (const float*)d_in[0];  // [16,1024]
  const float* h_prev  = (const float*)d_in[1];  // [16,2048]
  const float* W_ih_w  = (const float*)d_in[2];  // [2048,1024]
  const float* W_ih_b  = (const float*)d_in[3];  // [2048]
  const float* W_hh    = (const float*)d_in[4];  // [2048,2048]
  const float* temp    = (const float*)d_in[5];  // [1]
  const float* hh_mask = (const float*)d_in[6];  // [2048,2048]
  const float* u_noise = (const float*)d_in[7];  // [16,2048,2048]
  float* out = (float*)d_out;                    // [16,2048]
  float* ic  = (float*)d_ws;                     // [16,2048] scratch (128 KB)

  // Kernel 1: 128 WMMA tiles, 8 waves (256 thr) per block -> 16 blocks.
  ic_gemm_wmma<<<16, 256, 0, stream>>>(x_t, W_ih_w, W_ih_b, ic);

  // Kernel 2: one wave per (b,o) row: 16*2048 rows / 8 per block = 4096 blocks.
  gumbel_contrib<<<BATCH * 256, 256, 0, stream>>>(
      W_hh, hh_mask, u_noise, h_prev, ic, temp, out);
}